// DistillationLoss_32126355374570
// MI455X (gfx1250) — compile-verified
//
#include <hip/hip_runtime.h>

// ---------------------------------------------------------------------------
// DistillationLoss on MI455X (gfx1250, wave32)
//
// chamfer(a,b):  d2 = |a|^2 + |b|^2 - 2 a.b ; mean(min_row d) & mean(min_col d)
// The -2*a.b Gram term is computed 16x16-tile at a time with
// V_WMMA_F32_16X16X4_F32 (K=3 padded to 4, A pre-scaled by -2).
// Mins are tracked as d^2 (monotone) via u32-bit atomicMin (valid for >=0 f32),
// sqrt applied once per point in the finalize kernel.
//
// Round-1 -> Round-2 changes (from asm inspection):
//  * inner-loop B fragment now built from one unconditional 12-byte load
//    (global_load_b96) + v_cndmask selects -> no exec-masked load per tile
//  * column norm t2n computed lane-locally from the full point -> the
//    per-iteration ds_bpermute + s_wait_dscnt is gone
// ---------------------------------------------------------------------------

typedef float v2f __attribute__((ext_vector_type(2)));
typedef float v8f __attribute__((ext_vector_type(8)));

#define ALPHA_W 0.7f
#define LAMBDA_CHAMFER 1.0f
#define LAMBDA_EDGE 2.0f
#define FINF_BITS 0x7F800000u

// ---------------- workspace init: mins -> +inf bits, accumulators -> 0 ------
__global__ __launch_bounds__(256) void init_ws_kernel(unsigned int* ws,
                                                      int n_inf, int n_zero) {
  int i = blockIdx.x * blockDim.x + threadIdx.x;
  int stride = gridDim.x * blockDim.x;
  for (int j = i; j < n_inf; j += stride) ws[j] = FINF_BITS;
  for (int j = i; j < n_zero; j += stride) ws[n_inf + j] = 0u;
}

// ---------------- fused chamfer tile kernel --------------------------------
// One wave owns one 16-row tile of S and sweeps a chunk of 16-col tiles of T.
// Per tile: 1x v_wmma_f32_16x16x4_f32 + elementwise d2 + min reductions.
__global__ __launch_bounds__(256) void chamfer_wmma_kernel(
    const float* __restrict__ S, int NS,
    const float* __restrict__ T, int NT,
    unsigned int* __restrict__ rowmin,   // NS entries, f32-as-u32 min d^2
    unsigned int* __restrict__ colmin,   // NT entries, f32-as-u32 min d^2
    int chunkTiles) {
  const int lane = threadIdx.x & 31;
  const int wave = threadIdx.x >> 5;

  const int rowTiles = (NS + 15) >> 4;
  const int rowTile = blockIdx.x * 8 + wave;
  if (rowTile >= rowTiles) return;            // wave-uniform: EXEC stays full

  const int colTiles = (NT + 15) >> 4;
  const int ct0 = blockIdx.y * chunkTiles;
  const int ct1 = (ct0 + chunkTiles < colTiles) ? (ct0 + chunkTiles) : colTiles;
  if (ct0 >= ct1) return;

  const int h = lane >> 4;     // lane half (selects K pair for A/B, M half for D)
  const int l15 = lane & 15;

  // --- A fragment (16x4 f32): lane holds row (rowTile*16 + l15).
  //     Lanes 0-15 carry K=0,1 (x,y); lanes 16-31 carry K=2,3 (z,pad).
  //     Pre-scaled by -2 so the WMMA directly yields -2*dot.
  int mrow = rowTile * 16 + l15;
  if (mrow >= NS) mrow = NS - 1;               // clamp: duplicates are harmless
  {
    // (fragment built below from unconditional loads; no divergent branches)
  }
  float sxa = S[mrow * 3 + 0];
  float sya = S[mrow * 3 + 1];
  float sza = S[mrow * 3 + 2];
  v2f afrag;
  afrag.x = -2.0f * (h ? sza : sxa);
  afrag.y = h ? 0.0f : (-2.0f * sya);

  // --- |s|^2 for the 8 rows this lane's D fragment covers (rows h*8 + r)
  float s2r[8];
#pragma unroll
  for (int r = 0; r < 8; ++r) {
    int mr = rowTile * 16 + h * 8 + r;
    if (mr >= NS) mr = NS - 1;
    float x = S[mr * 3 + 0], y = S[mr * 3 + 1], z = S[mr * 3 + 2];
    s2r[r] = x * x + y * y + z * z;
  }

  float rowAcc[8];
#pragma unroll
  for (int r = 0; r < 8; ++r) rowAcc[r] = 3.0e38f;

  for (int ct = ct0; ct < ct1; ++ct) {
    int ncol = ct * 16 + l15;
    if (ncol >= NT) ncol = NT - 1;             // clamp: duplicate point, harmless
    // one contiguous 12B load per lane (both halves read the same column point)
    float tx = T[ncol * 3 + 0];
    float ty = T[ncol * 3 + 1];
    float tz = T[ncol * 3 + 2];
    float t2n = tx * tx + ty * ty + tz * tz;   // |t|^2, lane-local
    // B fragment (4x16 f32, mirror of A): lanes 0-15 K=0,1; lanes 16-31 K=2,3
    v2f bfrag;
    bfrag.x = h ? tz : tx;
    bfrag.y = h ? 0.0f : ty;

    v8f c = {0.f, 0.f, 0.f, 0.f, 0.f, 0.f, 0.f, 0.f};
    // D = A*B + C = -2 * (s . t) per 16x16 tile
    v8f d = __builtin_amdgcn_wmma_f32_16x16x4_f32(
        false, afrag, false, bfrag, (short)0, c, false, false);

    float colv = 3.0e38f;
#pragma unroll
    for (int r = 0; r < 8; ++r) {
      float d2 = s2r[r] + t2n + d[r];          // = |s|^2 + |t|^2 - 2 s.t
      d2 = fmaxf(d2, 0.0f);                    // clamp before bitwise min
      rowAcc[r] = fminf(rowAcc[r], d2);
      colv = fminf(colv, d2);
    }
    // column min: fold the two M-halves together, lanes 0..15 publish
    colv = fminf(colv, __shfl_xor(colv, 16, 32));
    if (lane < 16) atomicMin(&colmin[ncol], __float_as_uint(colv));
  }

  // row mins: reduce across the 16 lanes of each half (bit 4 untouched)
#pragma unroll
  for (int r = 0; r < 8; ++r) {
    float v = rowAcc[r];
    v = fminf(v, __shfl_xor(v, 1, 32));
    v = fminf(v, __shfl_xor(v, 2, 32));
    v = fminf(v, __shfl_xor(v, 4, 32));
    v = fminf(v, __shfl_xor(v, 8, 32));
    rowAcc[r] = v;
  }
  if (l15 == 0) {
#pragma unroll
    for (int r = 0; r < 8; ++r) {
      int mr = rowTile * 16 + h * 8 + r;
      if (mr < NS) atomicMin(&rowmin[mr], __float_as_uint(rowAcc[r]));
    }
  }
}

// ---------------- mesh edge loss: sum of squared edge lengths ---------------
__global__ __launch_bounds__(256) void edge_kernel(
    const float* __restrict__ V, const int* __restrict__ F, int NF,
    float* __restrict__ accum) {
  int i = blockIdx.x * blockDim.x + threadIdx.x;
  float s = 0.0f;
  if (i < NF) {
    int i0 = F[3 * i + 0], i1 = F[3 * i + 1], i2 = F[3 * i + 2];
    float ax = V[3 * i0], ay = V[3 * i0 + 1], az = V[3 * i0 + 2];
    float bx = V[3 * i1], by = V[3 * i1 + 1], bz = V[3 * i1 + 2];
    float cx = V[3 * i2], cy = V[3 * i2 + 1], cz = V[3 * i2 + 2];
    float e0x = ax - bx, e0y = ay - by, e0z = az - bz;
    float e1x = bx - cx, e1y = by - cy, e1z = bz - cz;
    float e2x = cx - ax, e2y = cy - ay, e2z = cz - az;
    s = e0x * e0x + e0y * e0y + e0z * e0z +
        e1x * e1x + e1y * e1y + e1z * e1z +
        e2x * e2x + e2y * e2y + e2z * e2z;
  }
#pragma unroll
  for (int m = 16; m >= 1; m >>= 1) s += __shfl_xor(s, m, 32);
  if ((threadIdx.x & 31) == 0) atomicAdd(accum, s);
}

// ---------------- finalize: means + sqrt + weighted combine -----------------
__global__ __launch_bounds__(256) void finalize_kernel(
    const unsigned int* __restrict__ rowmin_t, const unsigned int* __restrict__ rowmin_g, int NS,
    const unsigned int* __restrict__ colmin_t, int NT,
    const unsigned int* __restrict__ colmin_g, int NG,
    const float* __restrict__ edge_accum, int NF, float* __restrict__ out) {
  __shared__ float red0[256], red1[256], red2[256], red3[256];
  int tid = threadIdx.x;
  float s0 = 0.f, s1 = 0.f, s2 = 0.f, s3 = 0.f;
  for (int j = tid; j < NS; j += 256) {
    s0 += __builtin_amdgcn_sqrtf(fmaxf(__uint_as_float(rowmin_t[j]), 0.0f));
    s1 += __builtin_amdgcn_sqrtf(fmaxf(__uint_as_float(rowmin_g[j]), 0.0f));
  }
  for (int j = tid; j < NT; j += 256)
    s2 += __builtin_amdgcn_sqrtf(fmaxf(__uint_as_float(colmin_t[j]), 0.0f));
  for (int j = tid; j < NG; j += 256)
    s3 += __builtin_amdgcn_sqrtf(fmaxf(__uint_as_float(colmin_g[j]), 0.0f));
  red0[tid] = s0; red1[tid] = s1; red2[tid] = s2; red3[tid] = s3;
  __syncthreads();
  for (int off = 128; off > 0; off >>= 1) {
    if (tid < off) {
      red0[tid] += red0[tid + off];
      red1[tid] += red1[tid + off];
      red2[tid] += red2[tid + off];
      red3[tid] += red3[tid + off];
    }
    __syncthreads();
  }
  if (tid == 0) {
    float ch_t = 0.5f * (red0[0] / (float)NS + red2[0] / (float)NT);
    float ch_g = 0.5f * (red1[0] / (float)NS + red3[0] / (float)NG);
    float edge = edge_accum[0] / (3.0f * (float)NF);
    out[0] = LAMBDA_CHAMFER * (ALPHA_W * ch_t + (1.0f - ALPHA_W) * ch_g) +
             LAMBDA_EDGE * edge;
  }
}

// ---------------------------------------------------------------------------
extern "C" void kernel_launch(void* const* d_in, const int* in_sizes, int n_in,
                              void* d_out, int out_size, void* d_ws, size_t ws_size,
                              hipStream_t stream) {
  const float* S = (const float*)d_in[0];
  const float* T = (const float*)d_in[1];
  const float* G = (const float*)d_in[2];
  const int*   F = (const int*)d_in[3];
  const int NS = in_sizes[0] / 3;
  const int NT = in_sizes[1] / 3;
  const int NG = in_sizes[2] / 3;
  const int NF = in_sizes[3] / 3;

  // workspace layout (u32 words): rowmin_t | rowmin_g | colmin_t | colmin_g | edge_acc
  unsigned int* ws = (unsigned int*)d_ws;
  unsigned int* rowmin_t = ws;
  unsigned int* rowmin_g = ws + NS;
  unsigned int* colmin_t = ws + 2 * NS;
  unsigned int* colmin_g = ws + 2 * NS + NT;
  float* edge_acc = (float*)(ws + 2 * NS + NT + NG);

  const int n_inf = 2 * NS + NT + NG;
  init_ws_kernel<<<96, 256, 0, stream>>>(ws, n_inf, 1);

  const int rowTiles = (NS + 15) / 16;
  const int gx = (rowTiles + 7) / 8;           // 8 waves (row tiles) per block

  // teacher chamfer
  {
    const int colTiles = (NT + 15) / 16;
    const int chunkTiles = (colTiles + 4) / 5; // ~5 column chunks
    const int gy = (colTiles + chunkTiles - 1) / chunkTiles;
    chamfer_wmma_kernel<<<dim3(gx, gy), 256, 0, stream>>>(
        S, NS, T, NT, rowmin_t, colmin_t, chunkTiles);
  }
  // ground-truth chamfer
  {
    const int colTiles = (NG + 15) / 16;
    const int chunkTiles = (colTiles + 4) / 5;
    const int gy = (colTiles + chunkTiles - 1) / chunkTiles;
    chamfer_wmma_kernel<<<dim3(gx, gy), 256, 0, stream>>>(
        S, NS, G, NG, rowmin_g, colmin_g, chunkTiles);
  }

  edge_kernel<<<(NF + 255) / 256, 256, 0, stream>>>(S, F, NF, edge_acc);

  finalize_kernel<<<1, 256, 0, stream>>>(rowmin_t, rowmin_g, NS, colmin_t, NT,
                                         colmin_g, NG, edge_acc, NF,
                                         (float*)d_out);
}